// GCNReactionDirectionPredictor_52553219834611
// MI455X (gfx1250) — compile-verified
//
#include <hip/hip_runtime.h>

typedef __attribute__((ext_vector_type(2))) float v2f;
typedef __attribute__((ext_vector_type(8))) float v8f;

#define N_NODES  50000
#define N_EDGES  800000
#define FDIM     128
#define N_GRAPHS 512

// ---------------------------------------------------------------------------
// utility fills
// ---------------------------------------------------------------------------
__global__ void fill_f32(float* p, float v, int n) {
    int t = blockIdx.x * blockDim.x + threadIdx.x;
    if (t < n) p[t] = v;
}

// deg accumulate: deg[dst[e]] += 1  (deg pre-initialized to 1.0 for self loop)
__global__ void deg_accum(const int* __restrict__ dst, float* deg) {
    int e = blockIdx.x * blockDim.x + threadIdx.x;
    if (e < N_EDGES) atomicAdd(&deg[dst[e]], 1.0f);
}

__global__ void rsqrt_inplace(float* p, int n) {
    int t = blockIdx.x * blockDim.x + threadIdx.x;
    if (t < n) p[t] = rsqrtf(p[t]);
}

__global__ void edge_norm(const int* __restrict__ src, const int* __restrict__ dst,
                          const float* __restrict__ dinv, float* __restrict__ norm) {
    int e = blockIdx.x * blockDim.x + threadIdx.x;
    if (e < N_EDGES) norm[e] = dinv[src[e]] * dinv[dst[e]];
}

// embedding gather: x[n][c] = emb[features[n]][c]
__global__ void embed_gather(const int* __restrict__ feat, const float* __restrict__ emb,
                             float* __restrict__ x) {
    int t = blockIdx.x * blockDim.x + threadIdx.x;
    if (t >= N_NODES * FDIM) return;
    int n = t >> 7, c = t & 127;
    x[t] = emb[feat[n] * FDIM + c];
}

// ---------------------------------------------------------------------------
// H = X @ W   via V_WMMA_F32_16X16X4_F32, one wave per 16x16 tile.
// block = 256 threads = 8 waves; wave w handles N-tile w; blockIdx.x = M-tile.
// A frag (16x4): lanes 0-15 VGPR0/1 = A[M=lane][K=k..k+1], lanes 16-31 = K=k+2..k+3
// B frag (4x16): lanes 0-15 VGPR0/1 = B[K=k..k+1][N=lane], lanes 16-31 = K=k+2..k+3
// C/D: vgpr i, lane<16 -> (M=i, N=lane); lane>=16 -> (M=8+i, N=lane-16)
// ---------------------------------------------------------------------------
__global__ __launch_bounds__(256) void gemm_wmma_f32(const float* __restrict__ X,
                                                     const float* __restrict__ W,
                                                     float* __restrict__ H) {
    const int wave = threadIdx.x >> 5;   // N tile 0..7
    const int lane = threadIdx.x & 31;
    const int half = lane >> 4;          // 0: K pair 0, 1: K pair 1
    const int lr   = lane & 15;
    const int m0   = blockIdx.x * 16;
    const int n0   = wave * 16;

    const float* arow = X + (m0 + lr) * FDIM;   // row of A, this lane's M
    const float* bcol = W + n0 + lr;            // column of B, this lane's N

    v8f c = {};
#pragma unroll
    for (int k = 0; k < FDIM; k += 4) {
        const int ka = k + half * 2;
        v2f a, b;
        a.x = arow[ka];
        a.y = arow[ka + 1];
        b.x = bcol[ka * FDIM];
        b.y = bcol[(ka + 1) * FDIM];
        c = __builtin_amdgcn_wmma_f32_16x16x4_f32(
                /*neg_a=*/false, a, /*neg_b=*/false, b,
                /*c_mod=*/(short)0, c, /*reuse_a=*/false, /*reuse_b=*/false);
    }

    float* out = H + (m0 + half * 8) * FDIM + n0 + lr;
#pragma unroll
    for (int i = 0; i < 8; ++i) out[i * FDIM] = c[i];
}

// ---------------------------------------------------------------------------
// edge scatter: one wave32 per edge; lane handles 4 contiguous floats (float4
// gather from h[src], 4x global_atomic_add_f32 into agg[dst]).
// ---------------------------------------------------------------------------
__global__ __launch_bounds__(256) void scatter_edges(const int* __restrict__ src,
                                                     const int* __restrict__ dst,
                                                     const float* __restrict__ norm,
                                                     const float* __restrict__ H,
                                                     float* agg) {
    const int e = blockIdx.x * 8 + (threadIdx.x >> 5);
    if (e >= N_EDGES) return;
    const int lane = threadIdx.x & 31;
    const int s = src[e], d = dst[e];
    const float w = norm[e];
    const float4 v = ((const float4*)(H + s * FDIM))[lane];
    float* ad = agg + d * FDIM + lane * 4;
    atomicAdd(ad + 0, v.x * w);
    atomicAdd(ad + 1, v.y * w);
    atomicAdd(ad + 2, v.z * w);
    atomicAdd(ad + 3, v.w * w);
}

// x_out = relu(agg + h * dinv^2 + b)
__global__ void finalize_layer(const float* __restrict__ agg, const float* __restrict__ H,
                               const float* __restrict__ dinv, const float* __restrict__ bias,
                               float* __restrict__ Xout) {
    int t = blockIdx.x * blockDim.x + threadIdx.x;
    if (t >= N_NODES * FDIM) return;
    int n = t >> 7, c = t & 127;
    float di = dinv[n];
    float v = agg[t] + H[t] * di * di + bias[c];
    Xout[t] = v > 0.0f ? v : 0.0f;
}

// graph pooling: atomic accumulate sums and counts
__global__ void pool_accum(const float* __restrict__ X, const int* __restrict__ batch,
                           float* pooled, float* cnts) {
    int t = blockIdx.x * blockDim.x + threadIdx.x;
    if (t >= N_NODES * FDIM) return;
    int n = t >> 7, c = t & 127;
    int g = batch[n];
    atomicAdd(&pooled[g * FDIM + c], X[t]);
    if (c == 0) atomicAdd(&cnts[g], 1.0f);
}

// out[g] = dot(pooled[g]/max(cnt,1), fcW) + fcb
__global__ __launch_bounds__(128) void final_fc(const float* __restrict__ pooled,
                                                const float* __restrict__ cnts,
                                                const float* __restrict__ fcW,
                                                const float* __restrict__ fcb,
                                                float* __restrict__ out) {
    __shared__ float red[128];
    const int g = blockIdx.x;
    const int c = threadIdx.x;
    const float inv = 1.0f / fmaxf(cnts[g], 1.0f);
    red[c] = pooled[g * FDIM + c] * inv * fcW[c];
    __syncthreads();
#pragma unroll
    for (int s = 64; s > 0; s >>= 1) {
        if (c < s) red[c] += red[c + s];
        __syncthreads();
    }
    if (c == 0) out[g] = red[0] + fcb[0];
}

// ---------------------------------------------------------------------------
extern "C" void kernel_launch(void* const* d_in, const int* in_sizes, int n_in,
                              void* d_out, int out_size, void* d_ws, size_t ws_size,
                              hipStream_t stream) {
    const int*   edge_index = (const int*)d_in[0];      // [2, N_EDGES]
    const int*   features   = (const int*)d_in[1];      // [N_NODES, 1]
    const int*   batch      = (const int*)d_in[2];      // [N_NODES]
    const float* emb        = (const float*)d_in[3];    // [VOCAB, 128]
    const float* W1 = (const float*)d_in[4];
    const float* b1 = (const float*)d_in[5];
    const float* W2 = (const float*)d_in[6];
    const float* b2 = (const float*)d_in[7];
    const float* W3 = (const float*)d_in[8];
    const float* b3 = (const float*)d_in[9];
    const float* fcW = (const float*)d_in[10];          // [128, 1]
    const float* fcb = (const float*)d_in[11];          // [1]
    float* out = (float*)d_out;                         // [512]

    const int* src = edge_index;
    const int* dst = edge_index + N_EDGES;

    // workspace layout (floats)
    float* ws     = (float*)d_ws;
    float* x      = ws;                          // N*128
    float* h      = x + (size_t)N_NODES * FDIM;  // N*128
    float* agg    = h + (size_t)N_NODES * FDIM;  // N*128
    float* dinv   = agg + (size_t)N_NODES * FDIM;// N     (holds deg first)
    float* norm   = dinv + N_NODES;              // E
    float* pooled = norm + N_EDGES;              // 512*128
    float* cnts   = pooled + (size_t)N_GRAPHS * FDIM; // 512

    const int NF     = N_NODES * FDIM;
    const int nfBlk  = (NF + 255) / 256;
    const int eBlk   = (N_EDGES + 255) / 256;
    const int nBlk   = (N_NODES + 255) / 256;

    // degree + symmetric norm (once)
    fill_f32<<<nBlk, 256, 0, stream>>>(dinv, 1.0f, N_NODES);        // deg = 1 (self loop)
    deg_accum<<<eBlk, 256, 0, stream>>>(dst, dinv);
    rsqrt_inplace<<<nBlk, 256, 0, stream>>>(dinv, N_NODES);         // dinv = 1/sqrt(deg)
    edge_norm<<<eBlk, 256, 0, stream>>>(src, dst, dinv, norm);

    // embedding lookup
    embed_gather<<<nfBlk, 256, 0, stream>>>(features, emb, x);

    const float* Ws[3] = {W1, W2, W3};
    const float* bs[3] = {b1, b2, b3};
    for (int layer = 0; layer < 3; ++layer) {
        // h = x @ W  (WMMA f32 path); 50000/16 = 3125 M-tiles exactly
        gemm_wmma_f32<<<N_NODES / 16, 256, 0, stream>>>(x, Ws[layer], h);
        // agg = 0
        fill_f32<<<nfBlk, 256, 0, stream>>>(agg, 0.0f, NF);
        // agg[dst] += h[src] * norm   (one wave per edge)
        scatter_edges<<<N_EDGES / 8, 256, 0, stream>>>(src, dst, norm, h, agg);
        // x = relu(agg + h*dinv^2 + b)
        finalize_layer<<<nfBlk, 256, 0, stream>>>(agg, h, dinv, bs[layer], x);
    }

    // mean pool + fc
    fill_f32<<<(N_GRAPHS * FDIM + 255) / 256, 256, 0, stream>>>(pooled, 0.0f, N_GRAPHS * FDIM);
    fill_f32<<<(N_GRAPHS + 255) / 256, 256, 0, stream>>>(cnts, 0.0f, N_GRAPHS);
    pool_accum<<<nfBlk, 256, 0, stream>>>(x, batch, pooled, cnts);
    final_fc<<<N_GRAPHS, 128, 0, stream>>>(pooled, cnts, fcW, fcb, out);
}